// Decoder_90838558310722
// MI455X (gfx1250) — compile-verified
//
#include <hip/hip_runtime.h>
#include <cmath>

#define B_     64
#define S_     2048
#define EMB_   512
#define ENC2_  1024
#define DEC_   512
#define ATTN_  512
#define VOCAB_ 50257
#define CATK_  (DEC_ + ENC2_)   // 1536: attn_W inner dim, x_cat width
#define OUTK_  2048             // out_cat width, 4*DEC gate width

typedef __attribute__((ext_vector_type(2))) float v2f;
typedef __attribute__((ext_vector_type(8))) float v8f;

__device__ __forceinline__ v8f wmma_f32_4(v2f a, v2f b, v8f c) {
  // D = A(16x4,f32) * B(4x16,f32) + C(16x16,f32)
  return __builtin_amdgcn_wmma_f32_16x16x4_f32(false, a, false, b, (short)0, c,
                                               false, false);
}

__device__ __forceinline__ float sigm_(float x) { return 1.0f / (1.0f + expf(-x)); }

// ---------------------------------------------------------------------------
// Generic C[M,N] = A[M,K] * B[N,K]^T (+ bias) (+= existing C when accum)
// One 16x32 output slab (two 16x16 WMMA tiles sharing A) per wave.
// K must be a multiple of 32 (all call sites: 512/1536/2048).
// Loads are staged into distinct register arrays per 32-k chunk so the
// scheduler can issue them back-to-back and use graduated s_wait_loadcnt
// instead of a load->wait(0)->wmma serial chain.
// Out-of-range N lanes load a clamped valid row; their garbage stays confined
// to their own C column (never stored), so no masking math is needed.
// ---------------------------------------------------------------------------
__global__ void gemm_bias_wmma(const float* __restrict__ A, int lda,
                               const float* __restrict__ Bm, int ldb,
                               const float* __restrict__ bias,
                               float* __restrict__ C, int ldc,
                               int M, int N, int K, int accum) {
  int wave = (blockIdx.x * blockDim.x + threadIdx.x) >> 5;
  int lane = threadIdx.x & 31;
  int Mt = M >> 4;
  int Np = (N + 31) >> 5;                // pairs of 16-wide N tiles
  if (wave >= Mt * Np) return;           // wave-uniform exit: EXEC stays all-1s
  int mt = wave % Mt, np = wave / Mt;
  int m0 = mt << 4, n0 = np << 5;
  int t  = lane & 15;                    // A-row / B-row / C-col index in tile
  int kh = lane >> 4;                    // selects K pair within step of 4

  int  n[2]  = {n0 + t, n0 + 16 + t};
  bool nv[2] = {n[0] < N, n[1] < N};
  int  nb[2] = {nv[0] ? n[0] : (N - 1), nv[1] ? n[1] : (N - 1)};

  v8f acc[2];
#pragma unroll
  for (int j = 0; j < 2; ++j) {
    float bseed = (bias != nullptr) ? bias[nb[j]] : 0.0f;
#pragma unroll
    for (int r = 0; r < 8; ++r) {
      float cv = 0.0f;
      if (accum && nv[j]) cv = C[(size_t)(m0 + r + 8 * kh) * ldc + n[j]];
      acc[j][r] = cv + bseed;
    }
  }

  const float* ap  = A  + (size_t)(m0 + t) * lda + 2 * kh;
  const float* bp0 = Bm + (size_t)nb[0] * ldb + 2 * kh;
  const float* bp1 = Bm + (size_t)nb[1] * ldb + 2 * kh;
  for (int k0 = 0; k0 < K; k0 += 32) {
    __builtin_prefetch(bp0 + k0 + 256, 0, 0);   // global_prefetch on streamed W
    __builtin_prefetch(bp1 + k0 + 256, 0, 0);
    v2f av[8], b0[8], b1[8];                    // distinct regs -> batched loads
#pragma unroll
    for (int i = 0; i < 8; ++i) {
      av[i] = *(const v2f*)(ap  + k0 + 4 * i);
      b0[i] = *(const v2f*)(bp0 + k0 + 4 * i);
      b1[i] = *(const v2f*)(bp1 + k0 + 4 * i);
    }
#pragma unroll
    for (int i = 0; i < 8; ++i) {
      acc[0] = wmma_f32_4(av[i], b0[i], acc[0]);
      acc[1] = wmma_f32_4(av[i], b1[i], acc[1]);
    }
  }

#pragma unroll
  for (int j = 0; j < 2; ++j) {
#pragma unroll
    for (int r = 0; r < 8; ++r) {
      if (nv[j]) C[(size_t)(m0 + r + 8 * kh) * ldc + n[j]] = acc[j][r];
    }
  }
}

// ---------------------------------------------------------------------------
// Fused attention scores: one workgroup per s (32 waves).
// scores[b,s] = mask ? sum_a v[a]*tanh(hid_proj[b,a] + enc[s,b,:]·W_e[a,:])
// hid_proj already holds h_last·W_h^T + attn_b (WMMA C-init). ds_add_f32 reduce.
// ---------------------------------------------------------------------------
__global__ void attn_energy_kernel(const float* __restrict__ enc,
                                   const float* __restrict__ attn_W,
                                   const float* __restrict__ hid_proj,
                                   const float* __restrict__ v_w,
                                   const int*   __restrict__ mask,
                                   float* __restrict__ scores) {
  __shared__ float ssc[B_];
  int s = blockIdx.x;
  int tid = threadIdx.x;
  if (tid < B_) ssc[tid] = 0.0f;
  __syncthreads();

  int w = tid >> 5, lane = tid & 31;
  int mt = w & 3;                 // which 16 batch rows
  int ng = w >> 2;                // which group of 4 N-tiles (8 groups * 64 = 512)
  int m0 = mt << 4;
  int t = lane & 15, kh = lane >> 4;

  v8f acc[4];
#pragma unroll
  for (int j = 0; j < 4; ++j) {
    int n0 = ((ng << 2) + j) << 4;
#pragma unroll
    for (int r = 0; r < 8; ++r)
      acc[j][r] = hid_proj[(size_t)(m0 + r + 8 * kh) * ATTN_ + n0 + t];
  }

  const float* ap  = enc + ((size_t)s * B_ + m0 + t) * ENC2_ + 2 * kh;
  const float* bp0 = attn_W + ((size_t)((ng << 2) << 4) + t) * CATK_ + DEC_ + 2 * kh;
  for (int k0 = 0; k0 < ENC2_; k0 += 16) {
    v2f av[4], bv[4][4];                        // staged loads, distinct regs
#pragma unroll
    for (int i = 0; i < 4; ++i) {
      av[i] = *(const v2f*)(ap + k0 + 4 * i);
#pragma unroll
      for (int j = 0; j < 4; ++j)
        bv[i][j] = *(const v2f*)(bp0 + (size_t)j * 16 * CATK_ + k0 + 4 * i);
    }
#pragma unroll
    for (int i = 0; i < 4; ++i) {
#pragma unroll
      for (int j = 0; j < 4; ++j)
        acc[j] = wmma_f32_4(av[i], bv[i][j], acc[j]);
    }
  }

#pragma unroll
  for (int j = 0; j < 4; ++j) {
    int n0 = ((ng << 2) + j) << 4;
    float vv = v_w[n0 + t];
#pragma unroll
    for (int r = 0; r < 8; ++r) {
      atomicAdd(&ssc[m0 + r + 8 * kh], vv * tanhf(acc[j][r]));  // ds_add_f32
    }
  }
  __syncthreads();
  if (tid < B_) {
    float sc = ssc[tid];
    if (mask[(size_t)tid * S_ + s] == 0) sc = -1e10f;
    scores[(size_t)tid * S_ + s] = sc;
  }
}

// ---------------------------------------------------------------------------
__global__ void softmax_kernel(const float* __restrict__ scores,
                               float* __restrict__ attn) {
  __shared__ float red[256];
  int b = blockIdx.x, tid = threadIdx.x;
  const float* sc = scores + (size_t)b * S_;

  float mx = -3.0e38f;
  for (int s = tid; s < S_; s += 256) mx = fmaxf(mx, sc[s]);
  red[tid] = mx; __syncthreads();
  for (int off = 128; off > 0; off >>= 1) {
    if (tid < off) red[tid] = fmaxf(red[tid], red[tid + off]);
    __syncthreads();
  }
  mx = red[0]; __syncthreads();

  float sum = 0.0f;
  for (int s = tid; s < S_; s += 256) sum += expf(sc[s] - mx);
  red[tid] = sum; __syncthreads();
  for (int off = 128; off > 0; off >>= 1) {
    if (tid < off) red[tid] += red[tid + off];
    __syncthreads();
  }
  float inv = 1.0f / red[0];

  for (int s = tid; s < S_; s += 256)
    attn[(size_t)b * S_ + s] = expf(sc[s] - mx) * inv;
}

// ---------------------------------------------------------------------------
__global__ void embed_kernel(const int* __restrict__ tok,
                             const float* __restrict__ emb,
                             float* __restrict__ x_cat,
                             float* __restrict__ out_cat) {
  int idx = blockIdx.x * blockDim.x + threadIdx.x;   // B*EMB exactly
  int b = idx >> 9, e = idx & 511;
  float v = emb[(size_t)tok[b] * EMB_ + e];
  x_cat[b * CATK_ + e] = v;                          // x_cat[:, 0:512]
  out_cat[b * OUTK_ + (DEC_ + ENC2_) + e] = v;       // out_cat[:, 1536:2048]
}

// ---------------------------------------------------------------------------
__global__ void context_kernel(const float* __restrict__ attn,
                               const float* __restrict__ enc,
                               float* __restrict__ x_cat,
                               float* __restrict__ out_cat) {
  int idx = blockIdx.x * blockDim.x + threadIdx.x;   // B*ENC2 exactly
  int b = idx >> 10, h = idx & 1023;
  const float* ap = attn + (size_t)b * S_;
  float sum = 0.0f;
  for (int s = 0; s < S_; ++s)
    sum += ap[s] * enc[((size_t)s * B_ + b) * ENC2_ + h];
  x_cat[b * CATK_ + DEC_ + h] = sum;                 // x_cat[:, 512:1536]
  out_cat[b * OUTK_ + DEC_ + h] = sum;               // out_cat[:, 512:1536]
}

// ---------------------------------------------------------------------------
__global__ void lstm_cell_kernel(const float* __restrict__ gates,
                                 const float* __restrict__ c_prev,
                                 float* __restrict__ h_out,
                                 float* __restrict__ c_out,
                                 float* __restrict__ cat_out,
                                 int cat_ld, int cat_off) {
  int idx = blockIdx.x * blockDim.x + threadIdx.x;   // B*DEC exactly
  int b = idx >> 9, j = idx & 511;
  const float* g = gates + (size_t)b * (4 * DEC_);
  float gi = g[j];
  float gf = g[DEC_ + j];
  float gg = g[2 * DEC_ + j];
  float go = g[3 * DEC_ + j];
  float c = sigm_(gf) * c_prev[idx] + sigm_(gi) * tanhf(gg);
  float h = sigm_(go) * tanhf(c);
  c_out[idx] = c;
  h_out[idx] = h;
  if (cat_out != nullptr) cat_out[b * cat_ld + cat_off + j] = h;
}

// ---------------------------------------------------------------------------
extern "C" void kernel_launch(void* const* d_in, const int* in_sizes, int n_in,
                              void* d_out, int out_size, void* d_ws, size_t ws_size,
                              hipStream_t stream) {
  (void)in_sizes; (void)n_in; (void)out_size; (void)ws_size;
  const int*   tok    = (const int*)  d_in[0];
  const float* h0     = (const float*)d_in[1];   // [2,B,DEC]
  const float* c0     = (const float*)d_in[2];   // [2,B,DEC]
  const float* enc    = (const float*)d_in[3];   // [S,B,ENC2]
  const int*   mask   = (const int*)  d_in[4];   // [B,S]
  const float* emb    = (const float*)d_in[5];
  const float* attn_W = (const float*)d_in[6];   // [ATTN, DEC+ENC2]
  const float* attn_b = (const float*)d_in[7];
  const float* v_w    = (const float*)d_in[8];
  const float* W_ih0  = (const float*)d_in[9];   // [4DEC, 1536]
  const float* W_hh0  = (const float*)d_in[10];  // [4DEC, 512]
  const float* b_ih0  = (const float*)d_in[11];
  const float* b_hh0  = (const float*)d_in[12];
  const float* W_ih1  = (const float*)d_in[13];  // [4DEC, 512]
  const float* W_hh1  = (const float*)d_in[14];  // [4DEC, 512]
  const float* b_ih1  = (const float*)d_in[15];
  const float* b_hh1  = (const float*)d_in[16];
  const float* fc_W   = (const float*)d_in[17];  // [VOCAB, 2048]
  const float* fc_b   = (const float*)d_in[18];

  float* out   = (float*)d_out;                   // pred | h_new | c_new | attn
  float* pred  = out;
  float* h_new = out + (size_t)B_ * VOCAB_;
  float* c_new = h_new + 2 * B_ * DEC_;
  float* attn  = c_new + 2 * B_ * DEC_;

  float* ws       = (float*)d_ws;
  float* hid_proj = ws;                           // [B, ATTN]
  float* scores   = hid_proj + B_ * ATTN_;        // [B, S]
  float* x_cat    = scores + B_ * S_;             // [B, 1536]
  float* out_cat  = x_cat + B_ * CATK_;           // [B, 2048]
  float* gates0   = out_cat + B_ * OUTK_;         // [B, 2048]
  float* gates1   = gates0 + B_ * (4 * DEC_);     // [B, 2048]

  auto gemm = [&](const float* A, int lda, const float* Bm, int ldb,
                  const float* bias, float* C, int ldc,
                  int M, int N, int K, int accum) {
    int tiles  = (M >> 4) * ((N + 31) >> 5);      // one 16x32 slab per wave
    int blocks = (tiles + 7) / 8;                 // 8 waves per 256-thread block
    gemm_bias_wmma<<<blocks, 256, 0, stream>>>(A, lda, Bm, ldb, bias, C, ldc,
                                               M, N, K, accum);
  };

  // 1) embedding -> x_cat[:,0:512], out_cat[:,1536:2048]
  embed_kernel<<<(B_ * EMB_) / 256, 256, 0, stream>>>(tok, emb, x_cat, out_cat);

  // 2) hid_proj = h0[-1] · W_h^T + attn_b   (W_h = attn_W[:, 0:512])
  gemm(h0 + B_ * DEC_, DEC_, attn_W, CATK_, attn_b, hid_proj, ATTN_,
       B_, ATTN_, DEC_, 0);

  // 3) fused energy/tanh/v-dot -> masked scores[B,S]   (dominant 137 GF)
  attn_energy_kernel<<<S_, 1024, 0, stream>>>(enc, attn_W, hid_proj, v_w, mask,
                                              scores);

  // 4) softmax over S -> attn (directly into d_out slice)
  softmax_kernel<<<B_, 256, 0, stream>>>(scores, attn);

  // 5) context = attn · enc -> x_cat[:,512:1536], out_cat[:,512:1536]
  context_kernel<<<(B_ * ENC2_) / 256, 256, 0, stream>>>(attn, enc, x_cat, out_cat);

  // 6) LSTM layer 0
  gemm(x_cat, CATK_, W_ih0, CATK_, b_ih0, gates0, OUTK_, B_, 4 * DEC_, CATK_, 0);
  gemm(h0, DEC_, W_hh0, DEC_, b_hh0, gates0, OUTK_, B_, 4 * DEC_, DEC_, 1);
  lstm_cell_kernel<<<(B_ * DEC_) / 256, 256, 0, stream>>>(
      gates0, c0, h_new, c_new, nullptr, 0, 0);

  // 7) LSTM layer 1 (h1 read straight from d_out's h_new[0])
  gemm(h_new, DEC_, W_ih1, DEC_, b_ih1, gates1, OUTK_, B_, 4 * DEC_, DEC_, 0);
  gemm(h0 + B_ * DEC_, DEC_, W_hh1, DEC_, b_hh1, gates1, OUTK_, B_, 4 * DEC_, DEC_, 1);
  lstm_cell_kernel<<<(B_ * DEC_) / 256, 256, 0, stream>>>(
      gates1, c0 + B_ * DEC_, h_new + B_ * DEC_, c_new + B_ * DEC_,
      out_cat, OUTK_, 0);  // h2 -> out_cat[:, 0:512]

  // 8) prediction = out_cat · fc_W^T + fc_b   (streams 412 MB of fc_W once)
  gemm(out_cat, OUTK_, fc_W, OUTK_, fc_b, pred, VOCAB_, B_, VOCAB_, OUTK_, 0);
}